// RMILoss_20624432956244
// MI455X (gfx1250) — compile-verified
//
#include <hip/hip_runtime.h>
#include <hip/hip_bf16.h>
#include <math.h>

// ---------------------------------------------------------------------------
// RMI loss for MI455X (gfx1250, wave32).
// Bandwidth-bound problem (~210MB traffic, ~5 GFLOP) -> use f32 WMMA
// (V_WMMA_F32_16X16X4_F32, the highest-precision matrix path) for the 9x9
// covariance Grams, chunked into f64 accumulators to track the f64 reference.
// Gram kernel is branch-free per-lane: all control flow is wave-uniform, so
// EXEC is all-1s at every WMMA and loads pipeline without exec round-trips.
// ---------------------------------------------------------------------------

typedef __attribute__((ext_vector_type(2))) float v2f;
typedef __attribute__((ext_vector_type(8))) float v8f;

constexpr int   kN = 4, kC = 21, kH = 512, kW = 512;
constexpr int   kHW = kH * kW;
constexpr int   kNC = kN * kC;                 // 84
constexpr int   kPH = 171, kPW = 171;          // pooled H/W: floor((512+2-3)/3)+1
constexpr int   kPHW = kPH * kPW;
constexpr int   kNH = 169, kNW = 169;          // pooled minus (radius-1)
constexpr int   kL  = kNH * kNW;               // 28561 samples per (n,c)
constexpr float kClipMin = 1e-6f;
constexpr double kAlpha  = 5e-4;
constexpr int   kGramStride = 3 * 256;         // three 16x16 f64 tiles per (n,c)
constexpr int   kRowsPerWave = 8;              // f32 run length <= 8*169 samples
constexpr int   kBands = (kNH + kRowsPerWave - 1) / kRowsPerWave;  // 22
constexpr int   kWaves = kNC * kBands;         // 1848
constexpr int   kPad   = 16;                   // tail slack for overrun-safe loads

// ---------------- kernel 0: zero the f64 accumulator region ----------------
__global__ void zero_f64(double* p, int n) {
  for (int i = blockIdx.x * blockDim.x + threadIdx.x; i < n;
       i += gridDim.x * blockDim.x)
    p[i] = 0.0;
}

// ---------------- kernel 1: masked BCE-with-logits (sum) -------------------
__global__ void bce_kernel(const float* __restrict__ logits,
                           const long long* __restrict__ labels,
                           double* __restrict__ acc) {
  int p = blockIdx.x * blockDim.x + threadIdx.x;
  double sum = 0.0, cnt = 0.0;
  if (p < kN * kHW) {
    long long lab = labels[p];
    bool mask = lab < kC;
    if (mask) {
      int n = p / kHW, hw = p - n * kHW;
      const float* base = logits + (size_t)n * kC * kHW + hw;
      float s = 0.f;
      for (int c = 0; c < kC; ++c) {
        float x = base[(size_t)c * kHW];
        // logaddexp(0,x) - x*y, stable form
        float e = fmaxf(x, 0.f) + log1pf(expf(-fabsf(x)));
        if (c == (int)lab) e -= x;
        s += e;
      }
      sum = (double)s;
      cnt = 1.0;
    }
  }
  __shared__ double ss[256], sc[256];
  int t = threadIdx.x;
  ss[t] = sum; sc[t] = cnt;
  __syncthreads();
  for (int o = 128; o > 0; o >>= 1) {
    if (t < o) { ss[t] += ss[t + o]; sc[t] += sc[t + o]; }
    __syncthreads();
  }
  if (t == 0) { atomicAdd(&acc[0], ss[0]); atomicAdd(&acc[1], sc[0]); }
}

// ------- kernel 2: fused sigmoid/one-hot + 3x3 s3 p1 max-pool --------------
// Windows are disjoint (k==s), so every logit is read exactly once here.
__global__ void pool_kernel(const float* __restrict__ logits,
                            const long long* __restrict__ labels,
                            float* __restrict__ prPool,
                            unsigned char* __restrict__ laPool) {
  int idx = blockIdx.x * blockDim.x + threadIdx.x;
  if (idx >= kNC * kPHW) return;
  int nc = idx / kPHW, rem = idx - nc * kPHW;
  int ph = rem / kPW, pw = rem - ph * kPW;
  int n = nc / kC, c = nc - n * kC;
  const float* lg = logits + (size_t)nc * kHW;
  const long long* lb = labels + (size_t)n * kHW;
  float mpr = -1e30f;           // -inf padding: OOB taps skipped
  unsigned char mla = 0;
  for (int i = 0; i < 3; ++i) {
    int h = ph * 3 - 1 + i;
    if (h < 0 || h >= kH) continue;
    for (int j = 0; j < 3; ++j) {
      int w = pw * 3 - 1 + j;
      if (w < 0 || w >= kW) continue;
      int o = h * kW + w;
      long long lab = lb[o];
      float mk = (lab < kC) ? 1.f : 0.f;
      float x = lg[o];
      float pr = mk / (1.f + expf(-x)) + kClipMin;
      mpr = fmaxf(mpr, pr);
      if (mk > 0.f && (int)lab == c) mla = 1;
    }
  }
  prPool[idx] = mpr;
  laPool[idx] = mla;
}

// ------- kernel 3: WMMA Gram accumulation (row-based, branch-free) ---------
// A-tile rows: m<9 -> shifted-view (dy,dx) value, m==9 -> 1.0 (sum/count
// row), m>=10 -> 0.  Gram = A x A^T: the per-lane B register image of the
// transposed operand equals the A image, so the same regs feed both ports.
__global__ void gram_kernel(const float* __restrict__ prPool,
                            const unsigned char* __restrict__ laPool,
                            double* __restrict__ gram) {
  int wid = blockIdx.x * (blockDim.x >> 5) + (threadIdx.x >> 5);
  if (wid >= kWaves) return;                 // wave-uniform exit
  int nc   = wid / kBands;
  int band = wid - nc * kBands;
  int y0   = band * kRowsPerWave;
  int yEnd = (y0 + kRowsPerWave < kNH) ? y0 + kRowsPerWave : kNH;

  int lane  = threadIdx.x & 31;
  int m     = lane & 15;                     // A-matrix row (M)
  int khalf = lane >> 4;                     // K pair select (K=0,1 vs K=2,3)
  bool isDim = (m < 9);
  float oneV = (m == 9) ? 1.f : 0.f;         // ones row value, 0 for pad rows
  int dy = isDim ? m / 3 : 0;
  int dx = isDim ? m % 3 : 0;
  const float*         prPlane = prPool + (size_t)nc * kPHW;
  const unsigned char* laPlane = laPool + (size_t)nc * kPHW;

  v8f cLL = {}, cPP = {}, cLP = {};
  const int iters = (kNH + 3) >> 2;          // 43 K-steps per row
  for (int y = y0; y < yEnd; ++y) {          // wave-uniform trip count
    const float*         prRow = prPlane + (y + dy) * kPW + dx;
    const unsigned char* laRow = laPlane + (y + dy) * kPW + dx;
    if (y + 1 < yEnd) {                      // pull next row toward the WGP
      __builtin_prefetch(prRow + kPW, 0, 1); // -> global_prefetch_b8
      __builtin_prefetch(laRow + kPW, 0, 1);
    }
    for (int it = 0; it < iters; ++it) {
      int x0 = it * 4 + khalf * 2;
      // Unconditional loads (buffers tail-padded); validity via selects only,
      // so no exec divergence anywhere in this loop.
      float lb0 = (float)laRow[x0];
      float lb1 = (float)laRow[x0 + 1];
      float pb0 = prRow[x0];
      float pb1 = prRow[x0 + 1];
      bool v0 = (x0     < kNH);              // only it==42 is partially valid
      bool v1 = (x0 + 1 < kNH);
      float al0 = v0 ? (isDim ? lb0 : oneV) : 0.f;
      float al1 = v1 ? (isDim ? lb1 : oneV) : 0.f;
      float ap0 = v0 ? (isDim ? pb0 : oneV) : 0.f;
      float ap1 = v1 ? (isDim ? pb1 : oneV) : 0.f;
      v2f A = {al0, al1};
      v2f P = {ap0, ap1};
      // EXEC is all-1s here (control flow above is wave-uniform).
      cLL = __builtin_amdgcn_wmma_f32_16x16x4_f32(false, A, false, A,
                                                  (short)0, cLL, false, false);
      cPP = __builtin_amdgcn_wmma_f32_16x16x4_f32(false, P, false, P,
                                                  (short)0, cPP, false, false);
      cLP = __builtin_amdgcn_wmma_f32_16x16x4_f32(false, A, false, P,
                                                  (short)0, cLP, false, false);
    }
  }

  // Drain the f32 16x16 tiles into f64 accumulators (chunked accumulation
  // keeps f32 roundoff bounded; cross-band adds are f64 atomics).
  double* g = gram + (size_t)nc * kGramStride;
#pragma unroll
  for (int v = 0; v < 8; ++v) {
    int row = v + 8 * khalf;                 // C/D layout: M = vgpr + 8*lane[4]
    int col = m;                             // N = lane % 16
    int idx = row * 16 + col;
    atomicAdd(&g[idx],       (double)cLL[v]);
    atomicAdd(&g[256 + idx], (double)cPP[v]);
    atomicAdd(&g[512 + idx], (double)cLP[v]);
  }
}

// ------- kernel 4: per-(n,c) 9x9 f64 solve + final scalar ------------------
__global__ void solve_kernel(const double* __restrict__ gram,
                             const double* __restrict__ acc,
                             float* __restrict__ out) {
  __shared__ double sred[128];
  int t = threadIdx.x;
  double logdet = 0.0;
  if (t < kNC) {
    const double* g   = gram + (size_t)t * kGramStride;
    const double* Gll = g;
    const double* Gpp = g + 256;
    const double* Glp = g + 512;
    double Lc = Gll[9 * 16 + 9];             // sample count (= kL)
    double invL = 1.0 / Lc;
    double sl[9], sp[9];
    for (int d = 0; d < 9; ++d) { sl[d] = Gll[d * 16 + 9]; sp[d] = Gpp[d * 16 + 9]; }

    // Mp = pr_cov + alpha*I, Cholesky in place -> Lp
    double Mp[9][9];
    for (int i = 0; i < 9; ++i)
      for (int j = 0; j < 9; ++j)
        Mp[i][j] = Gpp[i * 16 + j] - sp[i] * sp[j] * invL + (i == j ? kAlpha : 0.0);
    for (int j = 0; j < 9; ++j) {
      double s = Mp[j][j];
      for (int k = 0; k < j; ++k) s -= Mp[j][k] * Mp[j][k];
      double d = sqrt(fmax(s, 1e-300));
      Mp[j][j] = d;
      double inv = 1.0 / d;
      for (int i = j + 1; i < 9; ++i) {
        double v = Mp[i][j];
        for (int k = 0; k < j; ++k) v -= Mp[i][k] * Mp[j][k];
        Mp[i][j] = v * inv;
      }
    }
    // Solve W * Lp^T = X  (X = cross-cov)  =>  X Mp^-1 X^T = W W^T
    double Wm[9][9];
    for (int i = 0; i < 9; ++i)
      for (int j = 0; j < 9; ++j) {
        double x = Glp[i * 16 + j] - sl[i] * sp[j] * invL;
        for (int k = 0; k < j; ++k) x -= Wm[i][k] * Mp[j][k];
        Wm[i][j] = x / Mp[j][j];
      }
    // appro = la_cov - W W^T + alpha*I; Cholesky log-det
    double Am[9][9];
    for (int i = 0; i < 9; ++i)
      for (int j = 0; j <= i; ++j) {
        double s = Gll[i * 16 + j] - sl[i] * sl[j] * invL;
        for (int k = 0; k < 9; ++k) s -= Wm[i][k] * Wm[j][k];
        Am[i][j] = s + (i == j ? kAlpha : 0.0);
      }
    for (int j = 0; j < 9; ++j) {
      double s = Am[j][j];
      for (int k = 0; k < j; ++k) s -= Am[j][k] * Am[j][k];
      double d = sqrt(fmax(s, 0.0));
      Am[j][j] = d;
      logdet += 2.0 * log(d + 1e-8);
      double inv = (d > 0.0) ? 1.0 / d : 0.0;
      for (int i = j + 1; i < 9; ++i) {
        double v = Am[i][j];
        for (int k = 0; k < j; ++k) v -= Am[i][k] * Am[j][k];
        Am[i][j] = v * inv;
      }
    }
  }
  sred[t] = logdet;
  __syncthreads();
  if (t == 0) {
    double r = 0.0;
    for (int i = 0; i < kNC; ++i) r += sred[i];
    double rmi = r / (double)(kN * 9);       // mean over n, /HALF_D, sum over c
    double bce = acc[0] / (acc[1] + 1.0);
    out[0] = (float)(0.5 * bce + 0.5 * rmi);
  }
}

// ---------------------------------------------------------------------------
extern "C" void kernel_launch(void* const* d_in, const int* in_sizes, int n_in,
                              void* d_out, int out_size, void* d_ws, size_t ws_size,
                              hipStream_t stream) {
  const float*     logits = (const float*)d_in[0];
  const long long* labels = (const long long*)d_in[1];   // int64 labels

  // ws layout: [2 f64 acc][84*768 f64 grams][pr_pool f32 +pad][la_pool u8 +pad]
  // (~13 MB); pads make the Gram kernel's <=3-element row overruns safe.
  double*        acc    = (double*)d_ws;
  double*        gram   = acc + 2;
  float*         prPool = (float*)(gram + (size_t)kNC * kGramStride);
  unsigned char* laPool = (unsigned char*)(prPool + (size_t)kNC * kPHW + kPad);

  int nZero = 2 + kNC * kGramStride;
  zero_f64<<<64, 256, 0, stream>>>(acc, nZero);
  bce_kernel<<<(kN * kHW + 255) / 256, 256, 0, stream>>>(logits, labels, acc);
  pool_kernel<<<(kNC * kPHW + 255) / 256, 256, 0, stream>>>(logits, labels,
                                                            prPool, laPool);
  int wavesPerBlock = 256 / 32;
  int gridG = (kWaves + wavesPerBlock - 1) / wavesPerBlock;
  gram_kernel<<<gridG, 256, 0, stream>>>(prPool, laPool, gram);
  solve_kernel<<<1, 128, 0, stream>>>(gram, acc, (float*)d_out);

  (void)in_sizes; (void)n_in; (void)out_size; (void)ws_size;
}